// DynamicConv1D_52725018526326
// MI455X (gfx1250) — compile-verified
//
#include <hip/hip_runtime.h>
#include <hip/hip_bf16.h>

typedef float v2f __attribute__((ext_vector_type(2)));
typedef float v4f __attribute__((ext_vector_type(4)));
typedef float v8f __attribute__((ext_vector_type(8)));

#define B_   16
#define L_   2048
#define D_   1024
#define NCH_ 8          // L-chunks for pooling stage 1
#define LC_  (L_ / NCH_)

// ---------------------------------------------------------------------------
// Stage 1: partial sums over L (deterministic two-stage mean, no atomics)
// grid (B, D/256, NCH), block 256; coalesced along D, streams x once.
// ---------------------------------------------------------------------------
__global__ __launch_bounds__(256) void pool_partial(const float* __restrict__ x,
                                                    float* __restrict__ partial) {
  const int b = blockIdx.x;
  const int d = blockIdx.y * 256 + threadIdx.x;
  const int c = blockIdx.z;
  const float* xb = x + ((size_t)b * L_ + (size_t)c * LC_) * D_ + d;
  float s = 0.0f;
#pragma unroll 4
  for (int l = 0; l < LC_; ++l) s += xb[(size_t)l * D_];
  partial[((size_t)b * NCH_ + c) * D_ + d] = s;
}

__global__ __launch_bounds__(256) void pool_finish(const float* __restrict__ partial,
                                                   float* __restrict__ pooled) {
  const int i = blockIdx.x * 256 + threadIdx.x;   // i = b*D + d
  const int b = i / D_;
  const int d = i - b * D_;
  float s = 0.0f;
#pragma unroll
  for (int c = 0; c < NCH_; ++c) s += partial[((size_t)b * NCH_ + c) * D_ + d];
  pooled[i] = s * (1.0f / (float)L_);
}

// ---------------------------------------------------------------------------
// 16xN GEMM with V_WMMA_F32_16X16X4_F32, fp32 end-to-end.
// One block (4 waves) per 16-wide N-tile; waves split K 4-ways, LDS-reduce.
// A 16x4 f32 layout: lane m=lane&15; v0: K = 2*(lane>>4), v1: K = +1.
// B 4x16 f32 layout: lane n=lane&15; same K striping.
// C/D 16x16 f32: vgpr r -> row r + 8*(lane>>4), col lane&15.
// ---------------------------------------------------------------------------
template <int N, bool GELU>
__global__ __launch_bounds__(128) void gemm16_wmma(const float* __restrict__ A,
                                                   const float* __restrict__ W,
                                                   const float* __restrict__ bias,
                                                   float* __restrict__ out,
                                                   int Kdim) {
  __shared__ float red[4][32][9];
  const int lane = threadIdx.x & 31;
  const int wave = threadIdx.x >> 5;
  const int half = lane >> 4;
  const int mn   = lane & 15;                  // A row (m) and B col (n) index
  const int n    = blockIdx.x * 16 + mn;

  const int kseg = wave * (Kdim >> 2);         // this wave's K segment
  const int kend = Kdim >> 2;

  const float* arow = A + (size_t)mn * Kdim + kseg + 2 * half;
  const float* bcol = W + (size_t)(kseg + 2 * half) * N + n;

  v8f acc = {};
#pragma unroll 8
  for (int k0 = 0; k0 < kend; k0 += 4) {
    v2f a = *(const v2f*)(arow + k0);          // K = k0+2h, k0+2h+1
    v2f b;
    b.x = bcol[(size_t)k0 * N];                // K = k0+2h
    b.y = bcol[(size_t)k0 * N + N];            // K = k0+2h+1
    __builtin_prefetch(bcol + (size_t)(k0 + 32) * N, 0, 3);  // global_prefetch_b8
    acc = __builtin_amdgcn_wmma_f32_16x16x4_f32(
        /*neg_a=*/false, a, /*neg_b=*/false, b,
        /*c_mod=*/(short)0, acc, /*reuse_a=*/false, /*reuse_b=*/false);
  }

#pragma unroll
  for (int r = 0; r < 8; ++r) red[wave][lane][r] = acc[r];
  __syncthreads();

  if (threadIdx.x < 32) {
    const float bv = bias[n];
#pragma unroll
    for (int r = 0; r < 8; ++r) {
      float v = red[0][lane][r] + red[1][lane][r] +
                red[2][lane][r] + red[3][lane][r] + bv;
      if (GELU) v = 0.5f * v * (1.0f + erff(v * 0.70710678118654752f));
      out[(size_t)(r + 8 * half) * N + n] = v;
    }
  }
}

// ---------------------------------------------------------------------------
// Depthwise dynamic conv1d, K=3, 'same' padding (cross-correlation).
// float4 along D (256 thr * 4 = 1024 = D); register-rolling over L:
// 1 x-load + 1 store per output after warm-up. x should hit L2 (128MB < 192MB).
// ---------------------------------------------------------------------------
#define LCHUNK_ 64
__global__ __launch_bounds__(256) void dynconv_k3(const float* __restrict__ x,
                                                  const float* __restrict__ ker,
                                                  float* __restrict__ out) {
  const int b  = blockIdx.x;
  const int l0 = blockIdx.y * LCHUNK_;
  const int d4 = threadIdx.x * 4;

  // per-sample taps: ker[b, d, k] at b*3072 + d*3 + k; 12 contiguous floats
  const float* kb = ker + (size_t)b * (D_ * 3) + (size_t)d4 * 3;
  v4f c0 = *(const v4f*)(kb);        // d0k0 d0k1 d0k2 d1k0
  v4f c1 = *(const v4f*)(kb + 4);    // d1k1 d1k2 d2k0 d2k1
  v4f c2 = *(const v4f*)(kb + 8);    // d2k2 d3k0 d3k1 d3k2
  v4f t0 = {c0.x, c0.w, c1.z, c2.y}; // tap 0 (x[l-1])
  v4f t1 = {c0.y, c1.x, c1.w, c2.z}; // tap 1 (x[l])
  v4f t2 = {c0.z, c1.y, c2.x, c2.w}; // tap 2 (x[l+1])

  const float* xb = x   + ((size_t)b * L_) * D_ + d4;
  float*       ob = out + ((size_t)b * L_) * D_ + d4;
  const v4f zero = {0.0f, 0.0f, 0.0f, 0.0f};

  v4f xm = (l0 > 0) ? *(const v4f*)(xb + (size_t)(l0 - 1) * D_) : zero;
  v4f xc = *(const v4f*)(xb + (size_t)l0 * D_);
#pragma unroll 4
  for (int l = l0; l < l0 + LCHUNK_; ++l) {
    v4f xp = (l + 1 < L_) ? *(const v4f*)(xb + (size_t)(l + 1) * D_) : zero;
    v4f o  = xm * t0 + xc * t1 + xp * t2;
    *(v4f*)(ob + (size_t)l * D_) = o;
    xm = xc;
    xc = xp;
  }
}

// ---------------------------------------------------------------------------
extern "C" void kernel_launch(void* const* d_in, const int* in_sizes, int n_in,
                              void* d_out, int out_size, void* d_ws, size_t ws_size,
                              hipStream_t stream) {
  const float* x  = (const float*)d_in[0];   // [16, 2048, 1024]
  const float* W1 = (const float*)d_in[1];   // [1024, 4096]
  const float* b1 = (const float*)d_in[2];   // [4096]
  const float* W2 = (const float*)d_in[3];   // [4096, 3072]
  const float* b2 = (const float*)d_in[4];   // [3072]
  float* out = (float*)d_out;                // [16, 2048, 1024]

  // workspace layout (floats)
  float* partial = (float*)d_ws;                       // 16*8*1024  = 131072
  float* pooled  = partial + (size_t)B_ * NCH_ * D_;   // 16*1024    =  16384
  float* h       = pooled  + (size_t)B_ * D_;          // 16*4096    =  65536
  float* kerw    = h       + (size_t)B_ * 4 * D_;      // 16*3072    =  49152

  // 1) mean pool over L (two-stage, deterministic)
  pool_partial<<<dim3(B_, D_ / 256, NCH_), 256, 0, stream>>>(x, partial);
  pool_finish<<<(B_ * D_) / 256, 256, 0, stream>>>(partial, pooled);

  // 2) h = gelu(pooled @ W1 + b1)   [16,1024]x[1024,4096]
  gemm16_wmma<4096, true><<<4096 / 16, 128, 0, stream>>>(pooled, W1, b1, h, 1024);

  // 3) ker = h @ W2 + b2            [16,4096]x[4096,3072]
  gemm16_wmma<3072, false><<<3072 / 16, 128, 0, stream>>>(h, W2, b2, kerw, 4096);

  // 4) depthwise dynamic conv, K=3, same padding
  dynconv_k3<<<dim3(B_, L_ / LCHUNK_), 256, 0, stream>>>(x, kerw, out);
}